// VolumeRenderer_29755533427099
// MI455X (gfx1250) — compile-verified
//
#include <hip/hip_runtime.h>
#include <math.h>

#define RES 128
#define VOLD 256
#define SLICE 65536   // 256*256
#define ROW   256

// ---------------------------------------------------------------------------
// L2 warming pass: pull the whole 64MB volume into the 192MB L2 via the
// gfx1250 prefetch path (global_prefetch_b8). 524288 cachelines of 128B.
// ---------------------------------------------------------------------------
__global__ void __launch_bounds__(256) warm_l2_kernel(const float* __restrict__ vol) {
    size_t line = (size_t)blockIdx.x * 256 + threadIdx.x;      // 0 .. 524287
    const char* p = (const char*)vol + line * 128;
    __builtin_prefetch(p, 0, 1);
}

// ---------------------------------------------------------------------------
// Ray-march kernel: one thread per pixel, 128 blocks x 128 threads
// (4 waves/block spread over 128 WGPs for chip-wide latency hiding).
// 4-step software pipeline -> 32 outstanding L2 gathers per wave.
// ---------------------------------------------------------------------------
__global__ void __launch_bounds__(128) volrend_kernel(const float* __restrict__ vol,
                                                      float* __restrict__ out,
                                                      int n_steps) {
    const int idx = blockIdx.x * 128 + threadIdx.x;   // 0..16383
    const int px  = idx & (RES - 1);
    const int py  = idx >> 7;

    // Camera basis (float32 values of the NumPy construction):
    const float cd  = -0.5773502691896258f;                     // -1/sqrt(3)
    const float srx =  0.7071067811865476f, srz = -0.7071067811865476f;
    const float sux = -0.4082482904638630f, suy = 0.8164965809277260f,
                suz = -0.4082482904638630f;

    const float xs = -0.5f + (float)px * (1.0f / 127.0f);
    const float ys =  0.5f - (float)py * (1.0f / 127.0f);

    float dx = cd + xs * srx + ys * sux;
    float dy = cd            + ys * suy;
    float dz = cd + xs * srz + ys * suz;
    const float invn = 1.0f / sqrtf(dx * dx + dy * dy + dz * dz);
    dx *= invn; dy *= invn; dz *= invn;

    const float dt = fminf(fminf(fabsf(1.0f / dx), fabsf(1.0f / dy)),
                           fabsf(1.0f / dz)) * (1.0f / 128.0f);

    const float ox = 3.0f, oy = 3.0f, oz = 3.0f;

    // Ray/box interval: outside |c| <= 1.00390625 every corner is invalid
    // (v == 0 -> state frozen). B has margin; marginal steps are handled
    // exactly by the validity-weighted sampler.
    const float B = 1.02f;
    float t0, t1;
    t0 = (-B - ox) / dx; t1 = (B - ox) / dx;
    float sEnter = fminf(t0, t1), sExit = fmaxf(t0, t1);
    t0 = (-B - oy) / dy; t1 = (B - oy) / dy;
    sEnter = fmaxf(sEnter, fminf(t0, t1)); sExit = fminf(sExit, fmaxf(t0, t1));
    t0 = (-B - oz) / dz; t1 = (B - oz) / dz;
    sEnter = fmaxf(sEnter, fminf(t0, t1)); sExit = fminf(sExit, fmaxf(t0, t1));
    if (sEnter > sExit) sExit = -1.0f;     // ray misses the volume entirely

    float alpha = 0.0f, rgb = 0.0f, s = 0.0f;
    int k = 0;

    // ---- Fast-forward to box entry with the EXACT fp add chain of the scan.
    while (k < n_steps && s < sEnter) { s += dt; ++k; }

    // Branchless trilinear sample: per-corner validity folded into weights,
    // indices clamped (identical result to reference g()).  Straight-line
    // code so four consecutive samples' 32 loads can be in flight together.
    auto sample = [&](float sv) -> float {
        const float cx = fmaf(dx, sv, ox);
        const float cy = fmaf(dy, sv, oy);
        const float cz = fmaf(dz, sv, oz);
        const float ix = ((cx + 1.0f) * 256.0f - 1.0f) * 0.5f;
        const float iy = ((cy + 1.0f) * 256.0f - 1.0f) * 0.5f;
        const float iz = ((cz + 1.0f) * 256.0f - 1.0f) * 0.5f;
        const float x0f = floorf(ix), y0f = floorf(iy), z0f = floorf(iz);
        const float tx = ix - x0f, ty = iy - y0f, tz = iz - z0f;
        const int x0 = (int)x0f, y0 = (int)y0f, z0 = (int)z0f;
        const int x1 = x0 + 1, y1 = y0 + 1, z1 = z0 + 1;

        const float wx0 = ((unsigned)x0 < 256u) ? (1.0f - tx) : 0.0f;
        const float wx1 = ((unsigned)x1 < 256u) ? tx          : 0.0f;
        const float wy0 = ((unsigned)y0 < 256u) ? (1.0f - ty) : 0.0f;
        const float wy1 = ((unsigned)y1 < 256u) ? ty          : 0.0f;
        const float wz0 = ((unsigned)z0 < 256u) ? (1.0f - tz) : 0.0f;
        const float wz1 = ((unsigned)z1 < 256u) ? tz          : 0.0f;

        const int x0c = min(max(x0, 0), 255), x1c = min(max(x1, 0), 255);
        const int y0c = min(max(y0, 0), 255), y1c = min(max(y1, 0), 255);
        const int z0c = min(max(z0, 0), 255), z1c = min(max(z1, 0), 255);

        const float* p00 = vol + ((z0c << 16) + (y0c << 8));
        const float* p01 = vol + ((z0c << 16) + (y1c << 8));
        const float* p10 = vol + ((z1c << 16) + (y0c << 8));
        const float* p11 = vol + ((z1c << 16) + (y1c << 8));

        const float v00 = wx0 * p00[x0c] + wx1 * p00[x1c];
        const float v01 = wx0 * p01[x0c] + wx1 * p01[x1c];
        const float v10 = wx0 * p10[x0c] + wx1 * p10[x1c];
        const float v11 = wx0 * p11[x0c] + wx1 * p11[x1c];
        return wz0 * (wy0 * v00 + wy1 * v01) + wz1 * (wy0 * v10 + wy1 * v11);
    };

    auto composite = [&](float v) {
        if (v > 0.1f) {                               // mask m
            const float dens = (1.0f - alpha) * v;    // density*(m folded in)
            rgb   = fmaf(v, dens, rgb);
            alpha = fminf(alpha + dens, 1.0f);
        }
    };

    while (k < n_steps) {
        // Wave-wide early exit: lanes with s > sExit or alpha == 1 are
        // provably frozen (density == 0 forever).
        if (__all((alpha >= 1.0f) || (s > sExit))) break;

        const float s0 = s;
        const float s1 = s0 + dt;      // exact reference add chain
        const float s2 = s1 + dt;
        const float s3 = s2 + dt;

        // Issue all four samples' loads together (32 outstanding gathers).
        // Addresses depend only on s, never on alpha -> pure reorder.
        const float v0 = sample(s0);
        const float v1 = sample(s1);
        const float v2 = sample(s2);
        const float v3 = sample(s3);

        // Prefetch the 4 cachelines of the sample ~8 steps ahead; the ray
        // crosses a y-row / z-slice cacheline almost every step.
        {
            const float sP = fmaf(dt, 8.0f, s);
            const int xp = min(max((int)floorf(fmaf(fmaf(dx, sP, ox) + 1.0f, 128.0f, -0.5f)), 0), 254);
            const int yp = min(max((int)floorf(fmaf(fmaf(dy, sP, oy) + 1.0f, 128.0f, -0.5f)), 0), 254);
            const int zp = min(max((int)floorf(fmaf(fmaf(dz, sP, oz) + 1.0f, 128.0f, -0.5f)), 0), 254);
            const float* q = vol + ((zp << 16) + (yp << 8) + xp);
            __builtin_prefetch(q,               0, 3);
            __builtin_prefetch(q + ROW,         0, 3);
            __builtin_prefetch(q + SLICE,       0, 3);
            __builtin_prefetch(q + SLICE + ROW, 0, 3);
        }

        // Composite strictly in scan order; guard tail steps past n_steps.
        composite(v0);
        if (k + 1 < n_steps) composite(v1);
        if (k + 2 < n_steps) composite(v2);
        if (k + 3 < n_steps) composite(v3);

        s = s3 + dt;          // exact add chain continues
        k += 4;
    }

    out[idx] = rgb;
}

// ---------------------------------------------------------------------------
// Host side: deterministically recompute n_steps = ceil(S_MAX / dt.max()).
// Pure CPU math -> graph-capture safe.
// ---------------------------------------------------------------------------
static int compute_n_steps() {
    const double cd  = -1.0 / sqrt(3.0);
    const double srx =  1.0 / sqrt(2.0), srz = -1.0 / sqrt(2.0);
    const double sux = -1.0 / sqrt(6.0), suy = 2.0 / sqrt(6.0),
                 suz = -1.0 / sqrt(6.0);
    double dtmax = 0.0;
    for (int j = 0; j < RES; ++j) {
        const double yy = 0.5 - (double)j / 127.0;
        for (int i = 0; i < RES; ++i) {
            const double xx = -0.5 + (double)i / 127.0;
            double dx = cd + xx * srx + yy * sux;
            double dy = cd            + yy * suy;
            double dz = cd + xx * srz + yy * suz;
            const double n = sqrt(dx * dx + dy * dy + dz * dz);
            dx /= n; dy /= n; dz /= n;
            const double mc = fmax(fabs(dx), fmax(fabs(dy), fabs(dz)));
            const double dt = 1.0 / (128.0 * mc);
            if (dt > dtmax) dtmax = dt;
        }
    }
    return (int)ceil(10.0 / dtmax);
}

extern "C" void kernel_launch(void* const* d_in, const int* in_sizes, int n_in,
                              void* d_out, int out_size, void* d_ws, size_t ws_size,
                              hipStream_t stream) {
    (void)in_sizes; (void)n_in; (void)out_size; (void)d_ws; (void)ws_size;
    const float* vol = (const float*)d_in[0];
    float* out = (float*)d_out;

    const int n_steps = compute_n_steps();

    // 64MB volume / 128B lines = 524288 prefetches -> hot 192MB L2.
    warm_l2_kernel<<<2048, 256, 0, stream>>>(vol);
    // 16384 rays, one thread each; 4 waves per block over 128 blocks.
    volrend_kernel<<<(RES * RES) / 128, 128, 0, stream>>>(vol, out, n_steps);
}